// GATLayer_25503515803792
// MI455X (gfx1250) — compile-verified
//
#include <hip/hip_runtime.h>
#include <hip/hip_bf16.h>
#include <math.h>

// ---------------------------------------------------------------------------
// GAT layer with hypernetwork weights, CDNA5 (gfx1250, wave32, WMMA f32,
// async global->LDS double buffering).
// B=8 S=64 N=256 STATE=128 IN_F=OUT_F=64
// ---------------------------------------------------------------------------

#define Bdim   8
#define Sdim   64
#define Ndim   256
#define STATE  128
#define Fdim   64            // IN_F == OUT_F
#define BS     (Bdim * Sdim) // 512
#define ALPHA  0.2f

typedef float v2f __attribute__((ext_vector_type(2)));
typedef float v8f __attribute__((ext_vector_type(8)));

// D = A(16x4, f32) * B(4x16, f32) + C(16x16, f32)
__device__ __forceinline__ v8f wmma_f32_16x16x4(v2f a, v2f b, v8f c) {
    return __builtin_amdgcn_wmma_f32_16x16x4_f32(
        /*neg_a=*/false, a, /*neg_b=*/false, b,
        /*c_mod=*/(short)0, c, /*reuse_a=*/false, /*reuse_b=*/false);
}

// Async global->LDS copy, 16B per lane, GVS addressing (saddr base + v offset).
// VDST carries the workgroup-relative LDS byte address (HW adds LDS_BASE).
__device__ __forceinline__ void async_load_b128(unsigned lds_byte_addr,
                                                unsigned gbyte_off,
                                                const float* gbase) {
    asm volatile("global_load_async_to_lds_b128 %0, %1, %2"
                 :
                 : "v"(lds_byte_addr), "v"(gbyte_off), "s"(gbase)
                 : "memory");
}
__device__ __forceinline__ void wait_async_le4() {
    asm volatile("s_wait_asynccnt 0x4" ::: "memory");
}
__device__ __forceinline__ void wait_async_le0() {
    asm volatile("s_wait_asynccnt 0x0" ::: "memory");
}

// ---------------------------------------------------------------------------
// Kernel 1: hypernetwork.  One block per (b,s).
//   hdn = relu(hyper@w1+b1); W = |hdn@w2+b2|  (64x64)
//   hda = relu(hyper@aw1+ab1); a = |hda@aw2+ab2| -> a1(64), a2(64)
// ---------------------------------------------------------------------------
__global__ void k1_hypernet(const float* __restrict__ hyper,
                            const float* __restrict__ w1, const float* __restrict__ b1,
                            const float* __restrict__ w2, const float* __restrict__ b2,
                            const float* __restrict__ aw1, const float* __restrict__ ab1,
                            const float* __restrict__ aw2, const float* __restrict__ ab2,
                            float* __restrict__ W_ws,
                            float* __restrict__ a1_ws, float* __restrict__ a2_ws) {
    const int bs = blockIdx.x;
    const int t  = threadIdx.x;

    __shared__ float s_h[STATE];
    __shared__ float s_hdn[32];
    __shared__ float s_hda[32];

    if (t < STATE) s_h[t] = hyper[bs * STATE + t];
    __syncthreads();

    if (t < 32) {
        float acc = b1[t];
        #pragma unroll 8
        for (int k = 0; k < STATE; ++k) acc += s_h[k] * w1[k * 32 + t];
        s_hdn[t] = fmaxf(acc, 0.0f);
    } else if (t < 64) {
        const int tt = t - 32;
        float acc = ab1[tt];
        #pragma unroll 8
        for (int k = 0; k < STATE; ++k) acc += s_h[k] * aw1[k * 32 + tt];
        s_hda[tt] = fmaxf(acc, 0.0f);
    }
    __syncthreads();

    // W: 4096 outputs, 16 per thread, coalesced across o.
    for (int r = 0; r < 16; ++r) {
        const int o = r * 256 + t;
        float acc = b2[o];
        #pragma unroll
        for (int k = 0; k < 32; ++k) acc += s_hdn[k] * w2[k * 4096 + o];
        W_ws[bs * 4096 + o] = fabsf(acc);
    }

    if (t < 128) {
        float acc = ab2[t];
        #pragma unroll
        for (int k = 0; k < 32; ++k) acc += s_hda[k] * aw2[k * 128 + t];
        const float v = fabsf(acc);
        if (t < 64) a1_ws[bs * 64 + t] = v;
        else        a2_ws[bs * 64 + (t - 64)] = v;
    }
}

// ---------------------------------------------------------------------------
// Kernel 2: h = inputs @ W  per (b,s):  (256x64)@(64x64) via WMMA f32 16x16x4.
// One block (8 waves) per (b,s).  W tile staged in LDS.
// ---------------------------------------------------------------------------
__global__ void k2_h_gemm(const float* __restrict__ inputs,
                          const float* __restrict__ W_ws,
                          float* __restrict__ h_ws) {
    const int bs   = blockIdx.x;
    const int t    = threadIdx.x;
    const int wave = t >> 5;
    const int lane = t & 31;
    const int lrow  = lane & 15;          // M/N index inside fragment
    const int khalf = (lane >> 4) << 1;   // 0 for lanes 0-15, 2 for lanes 16-31

    __shared__ float sW[64 * 64];
    #pragma unroll
    for (int r = 0; r < 16; ++r) sW[r * 256 + t] = W_ws[bs * 4096 + r * 256 + t];
    __syncthreads();

    const float* __restrict__ inB = inputs + (size_t)bs * Ndim * Fdim;
    float* __restrict__ hB        = h_ws   + (size_t)bs * Ndim * Fdim;

    // 64 C tiles (16 m-tiles x 4 n-tiles), 8 per wave.
    for (int tile = wave; tile < 64; tile += 8) {
        const int m0 = (tile >> 2) * 16;
        const int n0 = (tile & 3)  * 16;
        v8f c = {};
        #pragma unroll
        for (int k0 = 0; k0 < 64; k0 += 4) {
            // A fragment: rows of `inputs`, contiguous pairs along K.
            const v2f a = *(const v2f*)&inB[(m0 + lrow) * Fdim + k0 + khalf];
            // B fragment: columns of W (K rows striped like A).
            const int krow = k0 + khalf;
            v2f b;
            b.x = sW[krow * 64 + n0 + lrow];
            b.y = sW[(krow + 1) * 64 + n0 + lrow];
            c = wmma_f32_16x16x4(a, b, c);
        }
        const int rbase = m0 + ((lane >> 4) << 3);
        const int col   = n0 + lrow;
        #pragma unroll
        for (int r = 0; r < 8; ++r)
            hB[(rbase + r) * Fdim + col] = c[r];
    }
}

// ---------------------------------------------------------------------------
// Kernel 3: e1 = h@a1, e2 = h@a2.  One block per (b,s), one row (n) per thread.
// ---------------------------------------------------------------------------
__global__ void k3_edge(const float* __restrict__ h_ws,
                        const float* __restrict__ a1_ws, const float* __restrict__ a2_ws,
                        float* __restrict__ e1_ws, float* __restrict__ e2_ws) {
    const int bs = blockIdx.x;
    const int t  = threadIdx.x;
    __shared__ float sa1[64], sa2[64];
    if (t < 64)            sa1[t]      = a1_ws[bs * 64 + t];
    else if (t < 128)      sa2[t - 64] = a2_ws[bs * 64 + (t - 64)];
    __syncthreads();

    const float* __restrict__ hrow = h_ws + ((size_t)bs * Ndim + t) * Fdim;
    float acc1 = 0.0f, acc2 = 0.0f;
    #pragma unroll 8
    for (int o = 0; o < Fdim; ++o) {
        const float hv = hrow[o];
        acc1 += hv * sa1[o];
        acc2 += hv * sa2[o];
    }
    e1_ws[bs * Ndim + t] = acc1;
    e2_ws[bs * Ndim + t] = acc2;
}

// ---------------------------------------------------------------------------
// Kernel 4: softmax statistics over axis=1 (S).  For each (b,i,j):
//   M = max_s leaky(e1[b,s,i]+e2[b,s,j]);  Zinv = 1/sum_s exp(v - M)
// Masked entries (Adj<=0) never read these (uniform 1/S attention).
// One block per (b,i); j across 256 threads (e2 coalesced, e1 broadcast).
// ---------------------------------------------------------------------------
__global__ void k4_softmax_stats(const float* __restrict__ e1_ws,
                                 const float* __restrict__ e2_ws,
                                 float* __restrict__ M_ws, float* __restrict__ Zinv_ws) {
    const int b = blockIdx.x >> 8;
    const int i = blockIdx.x & 255;
    const int j = threadIdx.x;

    float m = -INFINITY, sum = 0.0f;
    #pragma unroll 4
    for (int s = 0; s < Sdim; ++s) {
        const int bs = b * Sdim + s;
        float v = e1_ws[bs * Ndim + i] + e2_ws[bs * Ndim + j];
        v = (v >= 0.0f) ? v : ALPHA * v;
        const float mn = fmaxf(m, v);
        sum = sum * __expf(m - mn) + __expf(v - mn);
        m = mn;
    }
    const int idx = (b * Ndim + i) * Ndim + j;
    M_ws[idx]    = m;
    Zinv_ws[idx] = 1.0f / sum;
}

// ---------------------------------------------------------------------------
// Kernel 5: h_prime = att @ h, out = elu(h_prime).
// Per (b,s): (256x256)@(256x64) via WMMA f32 16x16x4.
// h is double-buffered through LDS with GLOBAL_LOAD_ASYNC_TO_LDS_B128
// (ASYNCcnt): chunk p+1 streams in while WMMA consumes chunk p.
// Attention fragments computed on the fly (exp co-executes with WMMA).
// ---------------------------------------------------------------------------
__global__ void k5_aggregate(const float* __restrict__ h_ws,
                             const float* __restrict__ e1_ws, const float* __restrict__ e2_ws,
                             const int*   __restrict__ Adj,
                             const float* __restrict__ M_ws, const float* __restrict__ Zinv_ws,
                             float* __restrict__ out) {
    const int bs   = blockIdx.x;
    const int b    = bs >> 6;  // bs / Sdim
    const int t    = threadIdx.x;
    const int wave = t >> 5;
    const int lane = t & 31;
    const int lrow  = lane & 15;
    const int khalf = (lane >> 4) << 1;

    __shared__ float sh[2][64 * 64];   // double-buffered h chunk (2 x 16KB)
    __shared__ float se1[Ndim];
    __shared__ float se2[Ndim];
    se1[t] = e1_ws[bs * Ndim + t];
    se2[t] = e2_ws[bs * Ndim + t];

    const float* __restrict__ hB = h_ws + (size_t)bs * Ndim * Fdim;

    // Issue one 16KB chunk (64 j-rows x 64 o-cols) as 4 async b128 per thread,
    // fully coalesced: instruction q covers bytes [q*4096 + t*16, +16).
    auto issue_chunk = [&](int buf, int kc) {
        const unsigned ldsbase =
            (unsigned)(unsigned long long)(uintptr_t)&sh[buf][0];
        const float* gsrc = hB + kc * Fdim;
        #pragma unroll
        for (int q = 0; q < 4; ++q) {
            const unsigned off = (unsigned)((q * 256 + t) * 16);
            async_load_b128(ldsbase + off, off, gsrc);
        }
    };

    v8f c[2][4];
    #pragma unroll
    for (int it = 0; it < 2; ++it)
        #pragma unroll
        for (int ot = 0; ot < 4; ++ot)
            c[it][ot] = (v8f){};

    issue_chunk(0, 0);   // prologue: chunk 0 -> buf 0

    for (int p = 0; p < 4; ++p) {
        const int kc = p * 64;
        if (p < 3) {
            issue_chunk((p + 1) & 1, kc + 64);  // overlap next fetch with compute
            wait_async_le4();                   // oldest 4 (chunk p) have landed
        } else {
            wait_async_le0();
        }
        __syncthreads();   // all waves' chunk-p data visible (+ se1/se2 on p==0)

        const float* __restrict__ shp = sh[p & 1];

        #pragma unroll
        for (int it = 0; it < 2; ++it) {
            const int irow = (wave + it * 8) * 16 + lrow;   // global i of this lane
            const float e1i = se1[irow];
            const int adjRow = irow * Ndim;
            const int mzRow  = (b * Ndim + irow) * Ndim;

            // Attention A-fragments for this 64-wide K chunk (reused by 4 o-tiles).
            v2f af[16];
            #pragma unroll
            for (int kk = 0; kk < 16; ++kk) {
                const int j0 = kc + kk * 4 + khalf;
                #pragma unroll
                for (int p2 = 0; p2 < 2; ++p2) {
                    const int j = j0 + p2;
                    float x = e1i + se2[j];
                    x = (x >= 0.0f) ? x : ALPHA * x;
                    const float v = __expf(x - M_ws[mzRow + j]) * Zinv_ws[mzRow + j];
                    const float a = (Adj[adjRow + j] > 0) ? v : (1.0f / (float)Sdim);
                    af[kk][p2] = a;
                }
            }
            #pragma unroll
            for (int ot = 0; ot < 4; ++ot) {
                const int col = ot * 16 + lrow;
                v8f acc = c[it][ot];
                #pragma unroll
                for (int kk = 0; kk < 16; ++kk) {
                    const int krow = kk * 4 + khalf;
                    v2f bf;
                    bf.x = shp[krow * 64 + col];
                    bf.y = shp[(krow + 1) * 64 + col];
                    acc = wmma_f32_16x16x4(af[kk], bf, acc);
                }
                c[it][ot] = acc;
            }
        }
        __syncthreads();   // buf[p&1] free for the p+2 issue
    }

    // Epilogue: elu + store.
    float* __restrict__ outB = out + (size_t)bs * Ndim * Fdim;
    #pragma unroll
    for (int it = 0; it < 2; ++it) {
        const int rbase = (wave + it * 8) * 16 + ((lane >> 4) << 3);
        #pragma unroll
        for (int ot = 0; ot < 4; ++ot) {
            const int col = ot * 16 + lrow;
            #pragma unroll
            for (int r = 0; r < 8; ++r) {
                float v = c[it][ot][r];
                v = (v > 0.0f) ? v : expm1f(v);
                outB[(rbase + r) * Fdim + col] = v;
            }
        }
    }
}

// ---------------------------------------------------------------------------
// Launcher.  Workspace layout (floats):
//   W:     [0,        2097152)   (BS x 64 x 64)
//   h:     [2097152, 10485760)   (BS x 256 x 64)
//   a1:    [10485760, +32768)
//   a2:    [10518528, +32768)
//   e1:    [10551296, +131072)
//   e2:    [10682368, +131072)
//   M:     [10813440, +524288)
//   Zinv:  [11337728, +524288)   total ~47.4 MB
// ---------------------------------------------------------------------------
extern "C" void kernel_launch(void* const* d_in, const int* in_sizes, int n_in,
                              void* d_out, int out_size, void* d_ws, size_t ws_size,
                              hipStream_t stream) {
    const float* hyper = (const float*)d_in[0];
    const float* inputs = (const float*)d_in[1];
    const int*   Adj   = (const int*)d_in[2];
    const float* w1  = (const float*)d_in[3];
    const float* b1  = (const float*)d_in[4];
    const float* w2  = (const float*)d_in[5];
    const float* b2  = (const float*)d_in[6];
    const float* aw1 = (const float*)d_in[7];
    const float* ab1 = (const float*)d_in[8];
    const float* aw2 = (const float*)d_in[9];
    const float* ab2 = (const float*)d_in[10];
    float* out = (float*)d_out;

    float* ws = (float*)d_ws;
    float* W_ws    = ws;
    float* h_ws    = ws + 2097152;
    float* a1_ws   = ws + 10485760;
    float* a2_ws   = ws + 10518528;
    float* e1_ws   = ws + 10551296;
    float* e2_ws   = ws + 10682368;
    float* M_ws    = ws + 10813440;
    float* Zinv_ws = ws + 11337728;

    k1_hypernet<<<BS, 256, 0, stream>>>(hyper, w1, b1, w2, b2, aw1, ab1, aw2, ab2,
                                        W_ws, a1_ws, a2_ws);
    k2_h_gemm<<<BS, 256, 0, stream>>>(inputs, W_ws, h_ws);
    k3_edge<<<BS, 256, 0, stream>>>(h_ws, a1_ws, a2_ws, e1_ws, e2_ws);
    k4_softmax_stats<<<Bdim * Ndim, 256, 0, stream>>>(e1_ws, e2_ws, M_ws, Zinv_ws);
    k5_aggregate<<<BS, 256, 0, stream>>>(h_ws, e1_ws, e2_ws, Adj, M_ws, Zinv_ws, out);
}